// LLFullTrackMLObjectCondensation_35768487641760
// MI455X (gfx1250) — compile-verified
//
#include <hip/hip_runtime.h>
#include <math.h>

#define NPTS 150000
#define KOBJ 192
#define QMIN 0.5f
#define SB 1.0f

typedef float v2f __attribute__((ext_vector_type(2)));
typedef float v8f __attribute__((ext_vector_type(8)));

struct OCWS {
    unsigned long long alpha_packed[KOBJ]; // (beta_bits<<32) | (~index)
    float wsum[KOBJ];
    float pay0[KOBJ];
    float pay1[KOBJ];
    float qa[KOBJ];      // q[alpha_k] (0 if object absent)
    float anrm[KOBJ];    // |x_alpha|^2
    float acoord[KOBJ * 4]; // -2 * x_alpha, padded 4th = 0
    float att;
    float rep;
    float extra;         // noise + min_b + payload terms
    float noise_sum;
    float noise_cnt;
};

// ---------------------------------------------------------------- kernel 0
__global__ void oc_init(OCWS* __restrict__ w) {
    int i = blockIdx.x * blockDim.x + threadIdx.x;
    if (i < KOBJ) {
        w->alpha_packed[i] = 0ull;
        w->wsum[i] = 0.f;
        w->pay0[i] = 0.f;
        w->pay1[i] = 0.f;
    }
    if (i == 0) {
        w->att = 0.f; w->rep = 0.f; w->extra = 0.f;
        w->noise_sum = 0.f; w->noise_cnt = 0.f;
    }
}

// ---------------------------------------------------------------- kernel 1
// per-point scatter: q, |x|^2, per-object argmax beta / wsum / payload sums
__global__ void oc_scatter(const float* __restrict__ pb,
                           const float* __restrict__ cc,
                           const float* __restrict__ pe,
                           const float* __restrict__ pp,
                           const int*   __restrict__ tix,
                           const float* __restrict__ te,
                           const float* __restrict__ tp,
                           const float* __restrict__ tt,
                           OCWS* __restrict__ w,
                           float* __restrict__ qarr,
                           float* __restrict__ nrmarr) {
    int i = blockIdx.x * blockDim.x + threadIdx.x;
    if (i >= NPTS) return;

    float beta = fminf(fmaxf(pb[i], 1e-4f), 1.f - 1e-4f);
    float at = atanhf(beta);
    float wgt = at * at;
    float qi = (wgt + QMIN) * tt[i];
    qarr[i] = qi;

    float x0 = cc[3 * i + 0], x1 = cc[3 * i + 1], x2 = cc[3 * i + 2];
    nrmarr[i] = x0 * x0 + x1 * x1 + x2 * x2;

    int t = tix[i];
    if (t >= 0) {
        unsigned bb = __float_as_uint(beta);
        unsigned long long pack =
            ((unsigned long long)bb << 32) |
            (unsigned long long)(0xFFFFFFFFu - (unsigned)i);
        atomicMax(&w->alpha_packed[t], pack);
        atomicAdd(&w->wsum[t], wgt);

        float tei = te[i];
        float de = tei - pe[i];
        float el = de * de / (tei * tei);
        float t0 = tp[2 * i + 0], t1 = tp[2 * i + 1];
        float d0 = t0 - pp[2 * i + 0], d1 = t1 - pp[2 * i + 1];
        float pl = d0 * d0 / (t0 * t0) + d1 * d1 / (t1 * t1);
        atomicAdd(&w->pay0[t], wgt * el);
        atomicAdd(&w->pay1[t], wgt * pl);
    } else {
        atomicAdd(&w->noise_sum, beta);
        atomicAdd(&w->noise_cnt, 1.0f);
    }
}

// ---------------------------------------------------------------- kernel 2
// per-object finalize: gather alpha data, payload means, noise/min_b terms
__global__ void oc_objects(const float* __restrict__ pb,
                           const float* __restrict__ cc,
                           OCWS* __restrict__ w,
                           const float* __restrict__ qarr,
                           const float* __restrict__ nrmarr) {
    __shared__ float s_minb[256], s_cnt[256], s_p0[256], s_p1[256];
    int k = threadIdx.x;
    float minb = 0.f, cnt = 0.f, p0 = 0.f, p1 = 0.f;
    if (k < KOBJ) {
        unsigned long long pack = w->alpha_packed[k];
        if (pack != 0ull) {
            int idx = (int)(0xFFFFFFFFu - (unsigned)(pack & 0xFFFFFFFFull));
            float ba = fminf(fmaxf(pb[idx], 1e-4f), 1.f - 1e-4f);
            minb = 1.f - ba;
            cnt = 1.f;
            float ws_ = w->wsum[k] + 1e-9f;
            p0 = w->pay0[k] / ws_;
            p1 = w->pay1[k] / ws_;
            w->qa[k] = qarr[idx];
            w->anrm[k] = nrmarr[idx];
            w->acoord[4 * k + 0] = -2.f * cc[3 * idx + 0];
            w->acoord[4 * k + 1] = -2.f * cc[3 * idx + 1];
            w->acoord[4 * k + 2] = -2.f * cc[3 * idx + 2];
            w->acoord[4 * k + 3] = 0.f;
        } else {
            w->qa[k] = 0.f;
            w->anrm[k] = 0.f;
            w->acoord[4 * k + 0] = 0.f;
            w->acoord[4 * k + 1] = 0.f;
            w->acoord[4 * k + 2] = 0.f;
            w->acoord[4 * k + 3] = 0.f;
        }
    }
    s_minb[threadIdx.x] = minb;
    s_cnt[threadIdx.x] = cnt;
    s_p0[threadIdx.x] = p0;
    s_p1[threadIdx.x] = p1;
    __syncthreads();
    for (int off = 128; off > 0; off >>= 1) {
        if (threadIdx.x < off) {
            s_minb[threadIdx.x] += s_minb[threadIdx.x + off];
            s_cnt[threadIdx.x]  += s_cnt[threadIdx.x + off];
            s_p0[threadIdx.x]   += s_p0[threadIdx.x + off];
            s_p1[threadIdx.x]   += s_p1[threadIdx.x + off];
        }
        __syncthreads();
    }
    if (threadIdx.x == 0) {
        float nobj = fmaxf(s_cnt[0], 1.f);
        float nnoise = fmaxf(w->noise_cnt, 1.f);
        w->extra = SB * w->noise_sum / nnoise
                 + s_minb[0] / nobj
                 + (s_p0[0] + s_p1[0]) / nobj;
    }
}

// ---------------------------------------------------------------- kernel 3
// pairwise N x K pass via V_WMMA_F32_16X16X4_F32.
// A (16x4)  = x_i rows (z padded with 0), lanes 0-15:(x,y) lanes 16-31:(z,0)
// B (4x16)  = -2 * x_alpha columns, same K split across lane halves
// C (16x16) = |x_i|^2 + |x_alpha|^2   ==>  D = d^2 directly.
// Epilogue is fully branchless: hardware v_sqrt_f32 + selects, no exec-mask
// branching per element.
__global__ __launch_bounds__(256) void oc_pairs(const float* __restrict__ cc,
                                                const int*   __restrict__ tix,
                                                OCWS* __restrict__ w,
                                                const float* __restrict__ qarr,
                                                const float* __restrict__ nrmarr) {
    const int lane = threadIdx.x & 31;
    const int wave = threadIdx.x >> 5;
    const int rowbase = (blockIdx.x * 8 + wave) * 16;
    const int hi = lane >> 4;      // 0: K=0..1, 1: K=2..3
    const int col = lane & 15;

    // A tile (16x4 f32, 2 VGPRs per lane)
    v2f a; a.x = 0.f; a.y = 0.f;
    {
        int rA = rowbase + col;
        if (rA < NPTS) {
            if (hi == 0) { a.x = cc[3 * rA + 0]; a.y = cc[3 * rA + 1]; }
            else         { a.x = cc[3 * rA + 2]; a.y = 0.f; }
        }
    }

    // per-row data in the C/D layout (VGPR j <-> row rowbase+j+8*hi)
    float qrow[8]; int trow[8]; float nr[8];
#pragma unroll
    for (int j = 0; j < 8; ++j) {
        int r = rowbase + j + hi * 8;
        bool in = r < NPTS;
        qrow[j] = in ? qarr[r] : 0.f;
        trow[j] = in ? tix[r] : -2;
        nr[j]   = in ? nrmarr[r] : 0.f;
    }

    float att = 0.f, rep = 0.f;
#pragma unroll
    for (int t = 0; t < KOBJ / 16; ++t) {
        int n = t * 16 + col;
        v2f b;
        b.x = w->acoord[4 * n + 2 * hi + 0];
        b.y = w->acoord[4 * n + 2 * hi + 1];
        float qan = w->qa[n];      // 0 when object absent => kills both terms
        float an  = w->anrm[n];
        v8f c;
#pragma unroll
        for (int j = 0; j < 8; ++j) c[j] = nr[j] + an;

        // D = A * B + C = |x_i|^2 + |x_a|^2 - 2 x_i . x_a = d^2
        v8f d = __builtin_amdgcn_wmma_f32_16x16x4_f32(
            false, a, false, b, (short)0, c, false, false);

#pragma unroll
        for (int j = 0; j < 8; ++j) {
            float d2 = fmaxf(d[j], 0.f);
            float qq = qrow[j] * qan;
            // hardware sqrt (v_sqrt_f32): ~1 ulp, plenty for a loss hinge
            float s  = __builtin_amdgcn_sqrtf(d2 + 1e-9f);
            float hinge = fmaxf(0.f, 1.f - s);
            float m  = (trow[j] == n) ? 1.f : 0.f;   // member mask (branchless)
            att = fmaf(m * d2, qq, att);             // Mf * d2 * qq
            rep = fmaf((1.f - m) * hinge, qq, rep);  // (1-Mf) * relu * qq
        }
    }

    // wave32 butterfly reduction
#pragma unroll
    for (int m = 16; m > 0; m >>= 1) {
        att += __shfl_xor(att, m, 32);
        rep += __shfl_xor(rep, m, 32);
    }
    if (lane == 0 && rowbase < NPTS) {
        atomicAdd(&w->att, att);
        atomicAdd(&w->rep, rep);
    }
}

// ---------------------------------------------------------------- kernel 4
__global__ void oc_final(const OCWS* __restrict__ w, float* __restrict__ out) {
    out[0] = (w->att + w->rep) * (1.f / (float)NPTS) + w->extra;
}

// ---------------------------------------------------------------- launcher
extern "C" void kernel_launch(void* const* d_in, const int* in_sizes, int n_in,
                              void* d_out, int out_size, void* d_ws, size_t ws_size,
                              hipStream_t stream) {
    (void)in_sizes; (void)n_in; (void)out_size; (void)ws_size;
    const float* pb  = (const float*)d_in[0];  // pred_beta   [N,1]
    const float* cc  = (const float*)d_in[1];  // pred_ccoords[N,3]
    const float* pe  = (const float*)d_in[2];  // pred_energy [N,1]
    const float* pp  = (const float*)d_in[3];  // pred_pos    [N,2]
    // d_in[4] pred_time, d_in[5] pred_id unused by the loss
    const int*   tix = (const int*)  d_in[6];  // t_idx       [N,1]
    const float* te  = (const float*)d_in[7];  // t_energy    [N,1]
    const float* tp  = (const float*)d_in[8];  // t_pos       [N,2]
    const float* tt  = (const float*)d_in[9];  // t_time      [N,1]
    // t_pid, rowsplits unused (single event)

    OCWS*  w      = (OCWS*)d_ws;
    float* qarr   = (float*)((char*)d_ws + sizeof(OCWS));
    float* nrmarr = qarr + NPTS;

    oc_init<<<1, 256, 0, stream>>>(w);

    int nb = (NPTS + 255) / 256;
    oc_scatter<<<nb, 256, 0, stream>>>(pb, cc, pe, pp, tix, te, tp, tt, w, qarr, nrmarr);

    oc_objects<<<1, 256, 0, stream>>>(pb, cc, w, qarr, nrmarr);

    int waves  = (NPTS + 15) / 16;    // 9375 row tiles
    int blocks = (waves + 7) / 8;     // 8 waves (tiles) per 256-thread block
    oc_pairs<<<blocks, 256, 0, stream>>>(cc, tix, w, qarr, nrmarr);

    oc_final<<<1, 1, 0, stream>>>(w, (float*)d_out);
}